// GAT_27539330301988
// MI455X (gfx1250) — compile-verified
//
#include <hip/hip_runtime.h>
#include <hip/hip_bf16.h>
#include <float.h>

typedef __attribute__((ext_vector_type(2))) float v2f;
typedef __attribute__((ext_vector_type(8))) float v8f;

#define IN_CH 128
#define HID   5
#define H1    2
#define C1    10   /* H1*HID */
#define OUTC  10
#define EPS_BN 1e-5f

// ---------------------------------------------------------------------------
// Utility: fill a float buffer with a constant (for -FLT_MAX segment-max init)
// ---------------------------------------------------------------------------
__global__ void fill_k(float* __restrict__ p, long long n, float v) {
  long long i = (long long)blockIdx.x * blockDim.x + threadIdx.x;
  if (i < n) p[i] = v;
}

// ---------------------------------------------------------------------------
// Layer-1 GEMM: h = x @ W1   (x:[N,128], W1:[128,10] zero-padded to 16 cols)
// One wave32 per 16-node tile, V_WMMA_F32_16X16X4_F32, K stepped by 4.
// A layout: lane l -> m=l&15, k0=2*(l>>4); a={A[m][k0],A[m][k0+1]}
// B layout: lane l -> n=l&15, k0=2*(l>>4); b={B[k0][n],B[k0+1][n]}
// C/D     : c[r] = D[r + 8*(l>>4)][l&15]
// ---------------------------------------------------------------------------
__global__ __launch_bounds__(128) void gemm1_wmma(const float* __restrict__ x,
                                                  const float* __restrict__ W1,
                                                  float* __restrict__ hfeat, int N) {
  __shared__ float Wp[IN_CH * 16];
  for (int i = threadIdx.x; i < IN_CH * 16; i += 128) {
    int k = i >> 4, n = i & 15;
    Wp[i] = (n < C1) ? W1[k * C1 + n] : 0.0f;
  }
  __syncthreads();

  const int wave = threadIdx.x >> 5;
  const int lane = threadIdx.x & 31;
  const int mn   = lane & 15;        // row (for A) / col (for B,D)
  const int lh   = lane >> 4;        // lane-half selects K pair / M block
  const long long tile = (long long)blockIdx.x * 4 + wave;
  const long long base = tile * 16;
  long long row  = base + mn;
  long long rowc = row < N ? row : (long long)(N - 1);
  const float* xr = x + rowc * IN_CH + lh * 2;

  v8f c = {};
#pragma unroll
  for (int k = 0; k < IN_CH; k += 4) {
    v2f a, b;
    a.x = xr[k];
    a.y = xr[k + 1];
    b.x = Wp[(k + lh * 2) * 16 + mn];
    b.y = Wp[(k + lh * 2 + 1) * 16 + mn];
    c = __builtin_amdgcn_wmma_f32_16x16x4_f32(false, a, false, b, (short)0, c,
                                              false, false);
  }
  if (mn < C1) {
#pragma unroll
    for (int r = 0; r < 8; ++r) {
      long long node = base + r + 8 * lh;
      if (node < N) hfeat[node * C1 + mn] = c[r];
    }
  }
}

// ---------------------------------------------------------------------------
// Layer-2 GEMM: h2 = hbn @ W2  (hbn:[N,12] zero-padded, W2:[10,10] -> [12,16])
// ---------------------------------------------------------------------------
__global__ __launch_bounds__(128) void gemm2_wmma(const float* __restrict__ hbn,
                                                  const float* __restrict__ W2,
                                                  float* __restrict__ hfeat, int N) {
  __shared__ float Wp[12 * 16];
  for (int i = threadIdx.x; i < 12 * 16; i += 128) {
    int k = i >> 4, n = i & 15;
    Wp[i] = (k < OUTC && n < OUTC) ? W2[k * OUTC + n] : 0.0f;
  }
  __syncthreads();

  const int wave = threadIdx.x >> 5;
  const int lane = threadIdx.x & 31;
  const int mn   = lane & 15;
  const int lh   = lane >> 4;
  const long long tile = (long long)blockIdx.x * 4 + wave;
  const long long base = tile * 16;
  long long row  = base + mn;
  long long rowc = row < N ? row : (long long)(N - 1);
  const float* hr = hbn + rowc * 12 + lh * 2;

  v8f c = {};
#pragma unroll
  for (int k = 0; k < 12; k += 4) {
    v2f a, b;
    a.x = hr[k];
    a.y = hr[k + 1];
    b.x = Wp[(k + lh * 2) * 16 + mn];
    b.y = Wp[(k + lh * 2 + 1) * 16 + mn];
    c = __builtin_amdgcn_wmma_f32_16x16x4_f32(false, a, false, b, (short)0, c,
                                              false, false);
  }
  if (mn < OUTC) {
#pragma unroll
    for (int r = 0; r < 8; ++r) {
      long long node = base + r + 8 * lh;
      if (node < N) hfeat[node * OUTC + mn] = c[r];
    }
  }
}

// ---------------------------------------------------------------------------
// Per-node attention logits, layer 1 (2 heads x 5 ch) and layer 2 (1 head x 10)
// ---------------------------------------------------------------------------
__global__ void alphas1_k(const float* __restrict__ hfeat,
                          const float* __restrict__ a_src,
                          const float* __restrict__ a_dst,
                          float* __restrict__ as, float* __restrict__ ad, int N) {
  int i = blockIdx.x * blockDim.x + threadIdx.x;
  if (i >= N) return;
  const float* hv = hfeat + (long long)i * C1;
#pragma unroll
  for (int h = 0; h < H1; ++h) {
    float s = 0.f, d = 0.f;
#pragma unroll
    for (int cc = 0; cc < HID; ++cc) {
      float v = hv[h * HID + cc];
      s += v * a_src[h * HID + cc];
      d += v * a_dst[h * HID + cc];
    }
    as[i * H1 + h] = s;
    ad[i * H1 + h] = d;
  }
}

__global__ void alphas2_k(const float* __restrict__ hfeat,
                          const float* __restrict__ a_src,
                          const float* __restrict__ a_dst,
                          float* __restrict__ as, float* __restrict__ ad, int N) {
  int i = blockIdx.x * blockDim.x + threadIdx.x;
  if (i >= N) return;
  const float* hv = hfeat + (long long)i * OUTC;
  float s = 0.f, d = 0.f;
#pragma unroll
  for (int cc = 0; cc < OUTC; ++cc) {
    float v = hv[cc];
    s += v * a_src[cc];
    d += v * a_dst[cc];
  }
  as[i] = s;
  ad[i] = d;
}

// ---------------------------------------------------------------------------
// Edge passes. Edge id e < E uses edge_index; e >= E is the self loop e-E.
// ---------------------------------------------------------------------------
__global__ void edge_max1(const int* __restrict__ ei, const float* __restrict__ as,
                          const float* __restrict__ ad, float* __restrict__ emax,
                          long long E, int N) {
  long long e = (long long)blockIdx.x * blockDim.x + threadIdx.x;
  if (e >= E + N) return;
  int s, d;
  if (e < E) { s = ei[e]; d = ei[E + e]; } else { s = d = (int)(e - E); }
#pragma unroll
  for (int h = 0; h < H1; ++h) {
    float v = as[s * H1 + h] + ad[d * H1 + h];
    v = v > 0.f ? v : 0.2f * v;
    atomicMax(&emax[d * H1 + h], v);
  }
}

__global__ void edge_scatter1(const int* __restrict__ ei, const float* __restrict__ as,
                              const float* __restrict__ ad, const float* __restrict__ emax,
                              const float* __restrict__ hfeat, float* __restrict__ denom,
                              float* __restrict__ acc, long long E, int N) {
  long long e = (long long)blockIdx.x * blockDim.x + threadIdx.x;
  if (e >= E + N) return;
  int s, d;
  if (e < E) { s = ei[e]; d = ei[E + e]; } else { s = d = (int)(e - E); }
  const float* hs = hfeat + (long long)s * C1;
  float* ac = acc + (long long)d * C1;
#pragma unroll
  for (int h = 0; h < H1; ++h) {
    float v = as[s * H1 + h] + ad[d * H1 + h];
    v = v > 0.f ? v : 0.2f * v;
    float ex = __expf(v - emax[d * H1 + h]);
    atomicAdd(&denom[d * H1 + h], ex);
#pragma unroll
    for (int cc = 0; cc < HID; ++cc)
      atomicAdd(&ac[h * HID + cc], ex * hs[h * HID + cc]);
  }
}

__global__ void edge_max2(const int* __restrict__ ei, const float* __restrict__ as,
                          const float* __restrict__ ad, float* __restrict__ emax,
                          long long E, int N) {
  long long e = (long long)blockIdx.x * blockDim.x + threadIdx.x;
  if (e >= E + N) return;
  int s, d;
  if (e < E) { s = ei[e]; d = ei[E + e]; } else { s = d = (int)(e - E); }
  float v = as[s] + ad[d];
  v = v > 0.f ? v : 0.2f * v;
  atomicMax(&emax[d], v);
}

__global__ void edge_scatter2(const int* __restrict__ ei, const float* __restrict__ as,
                              const float* __restrict__ ad, const float* __restrict__ emax,
                              const float* __restrict__ hfeat, float* __restrict__ denom,
                              float* __restrict__ acc, long long E, int N) {
  long long e = (long long)blockIdx.x * blockDim.x + threadIdx.x;
  if (e >= E + N) return;
  int s, d;
  if (e < E) { s = ei[e]; d = ei[E + e]; } else { s = d = (int)(e - E); }
  float v = as[s] + ad[d];
  v = v > 0.f ? v : 0.2f * v;
  float ex = __expf(v - emax[d]);
  atomicAdd(&denom[d], ex);
  const float* hs = hfeat + (long long)s * OUTC;
  float* ac = acc + (long long)d * OUTC;
#pragma unroll
  for (int cc = 0; cc < OUTC; ++cc) atomicAdd(&ac[cc], ex * hs[cc]);
}

// ---------------------------------------------------------------------------
// Finalize layer 1 in place (acc <- acc/denom + b1) and reduce BN statistics
// (block partial sums in LDS, one global atomic per channel per block).
// ---------------------------------------------------------------------------
__global__ __launch_bounds__(256) void fin1_bnstats(float* __restrict__ acc,
                                                    const float* __restrict__ denom,
                                                    const float* __restrict__ b1,
                                                    float* __restrict__ bnsum, int N) {
  __shared__ float s0[C1], s1[C1];
  if (threadIdx.x < C1) { s0[threadIdx.x] = 0.f; s1[threadIdx.x] = 0.f; }
  __syncthreads();
  int i = blockIdx.x * 256 + threadIdx.x;
  if (i < N) {
#pragma unroll
    for (int j = 0; j < C1; ++j) {
      float o = acc[(long long)i * C1 + j] / (denom[i * H1 + j / HID] + 1e-16f) + b1[j];
      acc[(long long)i * C1 + j] = o;
      atomicAdd(&s0[j], o);
      atomicAdd(&s1[j], o * o);
    }
  }
  __syncthreads();
  if (threadIdx.x < C1) {
    atomicAdd(&bnsum[threadIdx.x], s0[threadIdx.x]);
    atomicAdd(&bnsum[16 + threadIdx.x], s1[threadIdx.x]);
  }
}

// BN (mean/var from sums) + ELU, writes zero-padded [N,12] for the K=12 GEMM2.
__global__ void bnapply_k(const float* __restrict__ h1out, const float* __restrict__ bnsum,
                          const float* __restrict__ gamma, const float* __restrict__ beta,
                          float* __restrict__ hbn, int N) {
  int i = blockIdx.x * blockDim.x + threadIdx.x;
  if (i >= N) return;
  float invN = 1.0f / (float)N;
#pragma unroll
  for (int j = 0; j < C1; ++j) {
    float mu  = bnsum[j] * invN;
    float var = bnsum[16 + j] * invN - mu * mu;
    float rs  = rsqrtf(var + EPS_BN);
    float v   = (h1out[(long long)i * C1 + j] - mu) * rs * gamma[j] + beta[j];
    v = v > 0.f ? v : expm1f(v);  // ELU(alpha=1)
    hbn[(long long)i * 12 + j] = v;
  }
  hbn[(long long)i * 12 + 10] = 0.f;
  hbn[(long long)i * 12 + 11] = 0.f;
}

__global__ void final_k(const float* __restrict__ acc, const float* __restrict__ denom,
                        const float* __restrict__ b2, float* __restrict__ out, int N) {
  int i = blockIdx.x * blockDim.x + threadIdx.x;
  if (i >= N) return;
  float inv = 1.0f / (denom[i] + 1e-16f);
#pragma unroll
  for (int cc = 0; cc < OUTC; ++cc)
    out[(long long)i * OUTC + cc] = acc[(long long)i * OUTC + cc] * inv + b2[cc];
}

// ---------------------------------------------------------------------------
extern "C" void kernel_launch(void* const* d_in, const int* in_sizes, int n_in,
                              void* d_out, int out_size, void* d_ws, size_t ws_size,
                              hipStream_t stream) {
  const float* x      = (const float*)d_in[0];
  const float* W1     = (const float*)d_in[1];
  const float* a_src1 = (const float*)d_in[2];
  const float* a_dst1 = (const float*)d_in[3];
  const float* b1     = (const float*)d_in[4];
  const float* gamma1 = (const float*)d_in[5];
  const float* beta1  = (const float*)d_in[6];
  const float* W2     = (const float*)d_in[7];
  const float* a_src2 = (const float*)d_in[8];
  const float* a_dst2 = (const float*)d_in[9];
  const float* b2     = (const float*)d_in[10];
  const int*   ei     = (const int*)d_in[11];

  const int       N = in_sizes[0] / IN_CH;
  const long long E = (long long)in_sizes[11] / 2;
  const long long Et = E + N;

  float* ws     = (float*)d_ws;
  float* emax1  = ws;                 // 2N
  float* denom1 = ws + 2LL * N;       // 2N   }
  float* acc1   = ws + 4LL * N;       // 10N  } contiguous zero region [2N,14N)
  float* emax2  = ws + 14LL * N;      // N
  float* denom2 = ws + 15LL * N;      // N    }
  float* acc2   = ws + 16LL * N;      // 10N  } contiguous zero region [15N,26N)
  float* as1    = ws + 26LL * N;      // 2N
  float* ad1    = ws + 28LL * N;      // 2N
  float* as2    = ws + 30LL * N;      // N
  float* ad2    = ws + 31LL * N;      // N
  float* hfeat1 = ws + 32LL * N;      // 10N (layer-1 pre-aggregation features)
  float* hbn    = ws + 42LL * N;      // 12N (BN+ELU output, K padded to 12)
  float* hfeat2 = ws + 54LL * N;      // 10N (layer-2 pre-aggregation features)
  float* bnsum  = ws + 64LL * N;      // 32  (10 sums @0, 10 sumsq @16)
  (void)ws_size; (void)n_in; (void)out_size;

  const int TB = 256;
  const int nodeBlocks = (N + TB - 1) / TB;
  const int edgeBlocks = (int)((Et + TB - 1) / TB);
  const int tiles      = (N + 15) / 16;
  const int gemmBlocks = (tiles + 3) / 4;   // 4 waves (tiles) per 128-thread block

  // ---- per-call re-init (deterministic; harness does not re-poison) ----
  hipMemsetAsync(denom1, 0, sizeof(float) * 12LL * N, stream);
  hipMemsetAsync(denom2, 0, sizeof(float) * 11LL * N, stream);
  hipMemsetAsync(bnsum,  0, sizeof(float) * 32, stream);
  fill_k<<<(int)((2LL * N + TB - 1) / TB), TB, 0, stream>>>(emax1, 2LL * N, -FLT_MAX);
  fill_k<<<nodeBlocks, TB, 0, stream>>>(emax2, N, -FLT_MAX);

  // ---- layer 1 ----
  gemm1_wmma<<<gemmBlocks, 128, 0, stream>>>(x, W1, hfeat1, N);
  alphas1_k<<<nodeBlocks, TB, 0, stream>>>(hfeat1, a_src1, a_dst1, as1, ad1, N);
  edge_max1<<<edgeBlocks, TB, 0, stream>>>(ei, as1, ad1, emax1, E, N);
  edge_scatter1<<<edgeBlocks, TB, 0, stream>>>(ei, as1, ad1, emax1, hfeat1,
                                               denom1, acc1, E, N);
  fin1_bnstats<<<nodeBlocks, TB, 0, stream>>>(acc1, denom1, b1, bnsum, N);
  bnapply_k<<<nodeBlocks, TB, 0, stream>>>(acc1, bnsum, gamma1, beta1, hbn, N);

  // ---- layer 2 ----
  gemm2_wmma<<<gemmBlocks, 128, 0, stream>>>(hbn, W2, hfeat2, N);
  alphas2_k<<<nodeBlocks, TB, 0, stream>>>(hfeat2, a_src2, a_dst2, as2, ad2, N);
  edge_max2<<<edgeBlocks, TB, 0, stream>>>(ei, as2, ad2, emax2, E, N);
  edge_scatter2<<<edgeBlocks, TB, 0, stream>>>(ei, as2, ad2, emax2, hfeat2,
                                               denom2, acc2, E, N);
  final_k<<<nodeBlocks, TB, 0, stream>>>(acc2, denom2, b2, (float*)d_out, N);
}